// GMMDense_25735444037921
// MI455X (gfx1250) — compile-verified
//
#include <hip/hip_runtime.h>

// GMM density: out[n,k] = normalized exp(-0.5*max((x-mu_k)^T Sinv_k (x-mu_k),0))
// N=65536, D=32, K=64.  fp32 end-to-end using V_WMMA_F32_16X16X4_F32.
// Round 3: fix async builtin signature — param is int4-vector pointer in
// addrspace(1)/(3), non-const.

typedef __attribute__((ext_vector_type(2))) float v2f;
typedef __attribute__((ext_vector_type(8))) float v8f;

#define GMM_N 65536
#define GMM_D 32
#define GMM_K 64

#if defined(__has_builtin)
#if __has_builtin(__builtin_amdgcn_global_load_async_to_lds_b128)
#define GMM_HAVE_ASYNC 1
#endif
#endif

#ifdef GMM_HAVE_ASYNC
typedef int gmm_v4i __attribute__((__vector_size__(4 * sizeof(int))));
__device__ __forceinline__ void gmm_async_copy16(const float* g, float* l) {
  __builtin_amdgcn_global_load_async_to_lds_b128(
      (__attribute__((address_space(1))) gmm_v4i*)g,
      (__attribute__((address_space(3))) gmm_v4i*)l,
      /*offset=*/0, /*cpol=*/0);
}
__device__ __forceinline__ void gmm_wait_async() {
#if __has_builtin(__builtin_amdgcn_s_wait_asynccnt)
  __builtin_amdgcn_s_wait_asynccnt(0);
#else
  asm volatile("s_wait_asynccnt 0" ::: "memory");
#endif
}
#endif

// ---------------------------------------------------------------------------
// Kernel 1: per-component Gauss-Jordan inversion of SPD 32x32 sigma, writing
// Sinv in WMMA B-fragment order:
//   flat idx = ((j*2 + t)*32 + lane)*2 + u   (j = K-step 0..7, t = col tile)
//   value    = Sinv[4*j + 2*(lane>>4) + u][16*t + (lane&15)]
// ---------------------------------------------------------------------------
__global__ __launch_bounds__(32) void gmm_invert_sigma(const float* __restrict__ sigma,
                                                       float* __restrict__ sfrag) {
  __shared__ __align__(16) float M[32 * 65];  // 32 rows x (A | I), stride 65 kills bank conflicts
  const int k = blockIdx.x;
  const int lane = threadIdx.x;

  const float* S = sigma + k * 1024;
  for (int i = lane; i < 1024; i += 32) {
    int r = i >> 5, c = i & 31;
    M[r * 65 + c] = S[i];
    M[r * 65 + 32 + c] = (r == c) ? 1.0f : 0.0f;
  }
  __syncthreads();

  for (int i = 0; i < 32; ++i) {
    // scale pivot row (lane = column, 2 columns each)
    float pinv = 1.0f / M[i * 65 + i];
    M[i * 65 + lane] *= pinv;
    M[i * 65 + 32 + lane] *= pinv;
    __syncthreads();
    // eliminate (lane = row)
    float f = M[lane * 65 + i];
    if (lane != i) {
      for (int c = 0; c < 64; ++c)
        M[lane * 65 + c] -= f * M[i * 65 + c];
    }
    __syncthreads();
  }

  // emit fragment-ordered Sinv
  for (int idx = lane; idx < 1024; idx += 32) {
    int u = idx & 1;
    int l = (idx >> 1) & 31;
    int pair = idx >> 6;            // j*2 + t
    int j = pair >> 1, t = pair & 1;
    int dk = 4 * j + 2 * (l >> 4) + u;
    int n = 16 * t + (l & 15);
    sfrag[k * 1024 + idx] = M[dk * 65 + 32 + n];
  }
}

// 3-compare / 7-select extraction of p[s] (avoids dynamic VGPR indexing)
__device__ __forceinline__ float sel8(const float p[8], int s) {
  float a0 = (s & 1) ? p[1] : p[0];
  float a1 = (s & 1) ? p[3] : p[2];
  float a2 = (s & 1) ? p[5] : p[4];
  float a3 = (s & 1) ? p[7] : p[6];
  float b0 = (s & 2) ? a1 : a0;
  float b1 = (s & 2) ? a3 : a2;
  return (s & 4) ? b1 : b0;
}

// ---------------------------------------------------------------------------
// Kernel 2: 8 waves/block, each wave owns a 16-row x tile; loop over K
// components with double-buffered Sinv fragments (async DMA prefetch of k+1
// overlapping the 16 v_wmma_f32_16x16x4_f32 of k), quad rows via shuffle
// butterfly, then exp + row-normalize epilogue.
// ---------------------------------------------------------------------------
__global__ __launch_bounds__(256) void gmm_forward(const float* __restrict__ x,
                                                   const float* __restrict__ mu,
                                                   const float* __restrict__ sfragG,
                                                   float* __restrict__ out) {
  __shared__ __align__(16) float s_mu[GMM_K * GMM_D];   // 8 KB
  __shared__ __align__(16) float s_x[8 * 16 * GMM_D];   // 16 KB (per-wave tiles)
  __shared__ __align__(16) float s_frag[2 * 1024];      // 8 KB double-buffered B frags
  __shared__ __align__(16) float s_q[8 * 16 * 65];      // quad tiles, stride 65

  const int tid = threadIdx.x;
  const int wave = tid >> 5;
  const int lane = tid & 31;
  const int hi = lane >> 4;       // half-wave index
  const int ln = lane & 15;

  const int rowBase = (blockIdx.x * 8 + wave) * 16;

  // stage mu (block-wide) and this wave's x tile (coalesced b128)
  for (int i = tid; i < GMM_K * GMM_D; i += 256) s_mu[i] = mu[i];
  float* xs = s_x + wave * 16 * GMM_D;
  {
    const float4* xg4 = (const float4*)(x + rowBase * GMM_D);
    float4* xs4 = (float4*)xs;
    for (int i = lane; i < 128; i += 32) xs4[i] = xg4[i];
  }
  // prologue: stage k=0 fragments into buffer 0
  ((float4*)s_frag)[tid] = ((const float4*)sfragG)[tid];
  __syncthreads();

  // Per-lane A-operand x values: xfrag[2j+u] = x[m][4j + 2*hi + u], m = ln
  float xfrag[16];
#pragma unroll
  for (int j = 0; j < 8; ++j) {
    xfrag[2 * j]     = xs[ln * GMM_D + 4 * j + 2 * hi];
    xfrag[2 * j + 1] = xs[ln * GMM_D + 4 * j + 2 * hi + 1];
  }
  // Per-lane column values for the contraction: rows (8*hi + j), col ln / ln+16
  float xc0[8], xc1[8];
#pragma unroll
  for (int j = 0; j < 8; ++j) {
    xc0[j] = xs[(8 * hi + j) * GMM_D + ln];
    xc1[j] = xs[(8 * hi + j) * GMM_D + 16 + ln];
  }

  float* qt = s_q + wave * 16 * 65;
  const v8f czero = {0.f, 0.f, 0.f, 0.f, 0.f, 0.f, 0.f, 0.f};

  for (int k = 0; k < GMM_K; ++k) {
    const int cur = k & 1;
    float* frag = s_frag + cur * 1024;
    float* fragNext = s_frag + (cur ^ 1) * 1024;
    const bool haveNext = (k + 1) < GMM_K;

    // kick off the k+1 fragment stage so it overlaps this iteration's math
#ifdef GMM_HAVE_ASYNC
    if (haveNext)
      gmm_async_copy16(sfragG + (k + 1) * 1024 + tid * 4, fragNext + tid * 4);
#else
    float4 tnext = make_float4(0.f, 0.f, 0.f, 0.f);
    if (haveNext) tnext = ((const float4*)(sfragG + (k + 1) * 1024))[tid];
#endif

    const float* muk = s_mu + k * GMM_D;
    const float muA = muk[ln];
    const float muB = muk[16 + ln];

    v8f c0 = czero, c1 = czero;
#pragma unroll
    for (int j = 0; j < 8; ++j) {
      v2f a;
      a[0] = xfrag[2 * j]     - muk[4 * j + 2 * hi];
      a[1] = xfrag[2 * j + 1] - muk[4 * j + 2 * hi + 1];
      v2f b0 = *(const v2f*)(frag + (2 * j) * 64 + lane * 2);
      v2f b1 = *(const v2f*)(frag + (2 * j + 1) * 64 + lane * 2);
      // D = A(16x4) * B(4x16) + C  (fp32 WMMA)
      c0 = __builtin_amdgcn_wmma_f32_16x16x4_f32(false, a, false, b0, (short)0, c0, false, false);
      c1 = __builtin_amdgcn_wmma_f32_16x16x4_f32(false, a, false, b1, (short)0, c1, false, false);
    }

    // quad[m] = sum_n T[m][n] * diff[m][n]; lane owns column (ln, ln+16),
    // rows 8*hi .. 8*hi+7 live in accumulator elements 0..7
    float p[8];
#pragma unroll
    for (int j = 0; j < 8; ++j)
      p[j] = c0[j] * (xc0[j] - muA) + c1[j] * (xc1[j] - muB);

    // butterfly sum across the 16 lanes of each half-wave
#pragma unroll
    for (int off = 1; off < 16; off <<= 1) {
#pragma unroll
      for (int j = 0; j < 8; ++j) p[j] += __shfl_xor(p[j], off, 16);
    }

    if (ln < 8) qt[(8 * hi + ln) * 65 + k] = sel8(p, ln);

    // finish the prefetch, then one barrier closes the iteration:
    // - makes fragNext visible to all waves for k+1
    // - guarantees all waves are done reading frag before it is overwritten
#ifdef GMM_HAVE_ASYNC
    if (haveNext) gmm_wait_async();
#else
    if (haveNext) ((float4*)fragNext)[tid] = tnext;
#endif
    __syncthreads();
  }

  // Epilogue: 2 lanes per row; exp(-0.5*max(q,0)), normalize over K=64, store b128
  const int m = lane >> 1, h = lane & 1;
  const float* qrow = qt + m * 65 + 32 * h;
  float e[32];
  float ssum = 0.f;
#pragma unroll
  for (int c = 0; c < 32; ++c) {
    float v = __expf(-0.5f * fmaxf(qrow[c], 0.f));
    e[c] = v;
    ssum += v;
  }
  ssum += __shfl_xor(ssum, 1, 32);  // combine the two half-row sums
  const float rinv = 1.0f / ssum;
  float* og = out + (rowBase + m) * GMM_K + 32 * h;
#pragma unroll
  for (int c = 0; c < 32; c += 4) {
    float4 v4 = make_float4(e[c] * rinv, e[c + 1] * rinv, e[c + 2] * rinv, e[c + 3] * rinv);
    *(float4*)(og + c) = v4;
  }
}

// ---------------------------------------------------------------------------
extern "C" void kernel_launch(void* const* d_in, const int* in_sizes, int n_in,
                              void* d_out, int out_size, void* d_ws, size_t ws_size,
                              hipStream_t stream) {
  (void)in_sizes; (void)n_in; (void)out_size; (void)ws_size;
  const float* x     = (const float*)d_in[0];  // (N, D)
  const float* mu    = (const float*)d_in[1];  // (K, D)
  const float* sigma = (const float*)d_in[2];  // (K, D, D)
  float* out = (float*)d_out;                  // (N, K)
  float* sfrag = (float*)d_ws;                 // K*1024 floats = 256 KB scratch

  gmm_invert_sigma<<<GMM_K, 32, 0, stream>>>(sigma, sfrag);
  gmm_forward<<<GMM_N / 128, 256, 0, stream>>>(x, mu, sfrag, out);
}